// GCN_71614284693720
// MI455X (gfx1250) — compile-verified
//
#include <hip/hip_runtime.h>
#include <hip/hip_bf16.h>

// GCN (3-layer, D=128) for MI455X / gfx1250, wave32.
// Roofline: GEMMs ~9.8 GFLOP (negligible); edge gather/scatter ~7.5 GB of
// irregular traffic -> bandwidth/atomic bound. fp32 WMMA (16x16x4) keeps
// full reference precision at zero roofline cost.

typedef __attribute__((ext_vector_type(2))) float v2f;
typedef __attribute__((ext_vector_type(4))) float v4f;
typedef __attribute__((ext_vector_type(8))) float v8f;

#define DF 128  // feature dim for all layers

// ---------------- small utility kernels ----------------

__global__ __launch_bounds__(256) void k_zero(float* __restrict__ p, int n) {
    int i = blockIdx.x * blockDim.x + threadIdx.x;
    if (i < n) p[i] = 0.0f;
}

__global__ __launch_bounds__(256) void k_degree(const int* __restrict__ dst,
                                                float* __restrict__ deg, int E) {
    int e = blockIdx.x * blockDim.x + threadIdx.x;
    if (e < E) atomicAdd(&deg[dst[e]], 1.0f);   // global_atomic_add_f32
}

__global__ __launch_bounds__(256) void k_dinv(float* __restrict__ deg, int n) {
    int i = blockIdx.x * blockDim.x + threadIdx.x;
    if (i < n) deg[i] = rsqrtf(deg[i] + 1.0f);
}

// ---------------- WMMA fp32 GEMM: H = act(X) @ W ; AGG = H*dinv^2 + b ----------------
// One wave computes a 16-row x 128-col strip: 8 accumulator tiles (64 VGPRs),
// 32 K-steps of v_wmma_f32_16x16x4_f32 per tile.
//
// W (128x128 fp32 = 64KB) is staged in LDS in the *exact per-lane B-operand
// layout*: float2 for (tile t, kstep k, lane l) lives at byte offset
//     l*8 + (t*32 + k)*256
// so every inner-loop fetch is a single ds_load_b64 from one base VGPR with a
// compile-time immediate offset (no per-load VALU address math), and for a
// fixed (t,k) the 32 lanes read 256 consecutive bytes = all 64 LDS banks
// exactly once (conflict-free). Address-independent loads let the scheduler
// hoist ds_loads and overlap LDS latency with the WMMA pipe.
template <bool RELU_IN>
__global__ __launch_bounds__(256) void k_gemm_wmma(
        const float* __restrict__ X, const float* __restrict__ W,
        const float* __restrict__ bias, const float* __restrict__ dinv,
        float* __restrict__ H, float* __restrict__ AGG,
        int strips, int nrows) {
    __shared__ __align__(16) float ldsW[DF * DF];   // 64 KB

    // Stage W[kk][n] -> lds[(t*32+k)*64 + lane*2 + lo] (dwords), where
    // t=n>>4, nloc=n&15, k=kk>>2, khalf=(kk>>1)&1, lo=kk&1, lane=khalf*16+nloc.
    for (int idx = threadIdx.x; idx < DF * DF; idx += blockDim.x) {
        int kk = idx >> 7;            // row of W (K index)
        int n  = idx & 127;           // col of W (N index)
        int t     = n >> 4;
        int nloc  = n & 15;
        int k     = kk >> 2;
        int khalf = (kk >> 1) & 1;
        int lo    = kk & 1;
        ldsW[((t << 5) + k) * 64 + (khalf * 16 + nloc) * 2 + lo] = W[idx];
    }
    __syncthreads();

    const int wave  = threadIdx.x >> 5;
    const int lane  = threadIdx.x & 31;
    const int strip = blockIdx.x * (blockDim.x >> 5) + wave;
    if (strip >= strips) return;                 // wave-uniform: EXEC stays all-ones

    const int nloc  = lane & 15;                 // A: row-in-strip / B,D: col-in-tile
    const int khalf = lane >> 4;                 // which K half-pair this lane owns

    // A operand: lane reads float2 from row (strip*16 + nloc), cols 4k + 2*khalf.
    int arow_i = strip * 16 + nloc;
    if (arow_i >= nrows) arow_i = nrows - 1;     // clamp (pad rows never stored)
    const float* __restrict__ arow = X + (size_t)arow_i * DF + 2 * khalf;

    // B operand base: one VGPR; every (t,k) fetch is an immediate DS offset.
    const v2f* __restrict__ bbase = (const v2f*)ldsW + lane;

    v8f acc[8] = {};
#pragma unroll
    for (int k = 0; k < 32; ++k) {
        v2f a = *(const v2f*)(arow + 4 * k);     // global_load_b64, imm offset
        if (RELU_IN) {
            a.x = fmaxf(a.x, 0.0f);
            a.y = fmaxf(a.y, 0.0f);
        }
#pragma unroll
        for (int t = 0; t < 8; ++t) {
            v2f b = bbase[(t * 32 + k) * 32];    // ds_load_b64, imm offset
            acc[t] = __builtin_amdgcn_wmma_f32_16x16x4_f32(
                         false, a, false, b, (short)0, acc[t], false, false);
        }
    }

    // Epilogue: lane l, acc[t] elem r -> D[row = strip*16 + r + 8*khalf][col = 16t + nloc]
    float bcol[8];
#pragma unroll
    for (int t = 0; t < 8; ++t) bcol[t] = bias[t * 16 + nloc];

    const int rbase = strip * 16 + 8 * khalf;
#pragma unroll
    for (int r = 0; r < 8; ++r) {
        int row = rbase + r;
        if (row >= nrows) continue;
        float dv = dinv[row];
        float d2 = dv * dv;
        size_t ro = (size_t)row * DF;
#pragma unroll
        for (int t = 0; t < 8; ++t) {
            float val = acc[t][r];
            int col = t * 16 + nloc;
            H[ro + col]   = val;                        // pre-aggregation features
            AGG[ro + col] = fmaf(val, d2, bcol[t]);     // self-loop + bias init
        }
    }
}

// ---------------- edge scatter: AGG[dst] += H[src] * dinv[src]*dinv[dst] ----------------
// One wave per edge; each lane owns a float4 of the 128-float feature row.
__global__ __launch_bounds__(256) void k_scatter(
        const float* __restrict__ H, const int* __restrict__ src,
        const int* __restrict__ dst, const float* __restrict__ dinv,
        float* __restrict__ AGG, int E) {
    int e = blockIdx.x * (blockDim.x >> 5) + (threadIdx.x >> 5);
    if (e >= E) return;
    int lane = threadIdx.x & 31;
    int s = src[e], d = dst[e];
    float nrm = dinv[s] * dinv[d];
    v4f v = *(const v4f*)(H + (size_t)s * DF + lane * 4);   // coalesced row gather
    float* out = AGG + (size_t)d * DF + lane * 4;
    atomicAdd(out + 0, v.x * nrm);
    atomicAdd(out + 1, v.y * nrm);
    atomicAdd(out + 2, v.z * nrm);
    atomicAdd(out + 3, v.w * nrm);
}

// ---------------- final row L2 normalize (wave per row) ----------------
__global__ __launch_bounds__(256) void k_normalize(const float* __restrict__ in,
                                                   float* __restrict__ out, int n) {
    int row = blockIdx.x * (blockDim.x >> 5) + (threadIdx.x >> 5);
    if (row >= n) return;
    int lane = threadIdx.x & 31;
    v4f v = *(const v4f*)(in + (size_t)row * DF + lane * 4);
    float ss = v.x * v.x + v.y * v.y + v.z * v.z + v.w * v.w;
#pragma unroll
    for (int m = 16; m > 0; m >>= 1) ss += __shfl_xor(ss, m, 32);
    float sc = 1.0f / fmaxf(sqrtf(ss), 1e-12f);
    v4f o = {v.x * sc, v.y * sc, v.z * sc, v.w * sc};
    *(v4f*)(out + (size_t)row * DF + lane * 4) = o;
}

// ---------------- orchestration ----------------

extern "C" void kernel_launch(void* const* d_in, const int* in_sizes, int n_in,
                              void* d_out, int out_size, void* d_ws, size_t ws_size,
                              hipStream_t stream) {
    const float* x  = (const float*)d_in[0];
    const int*   ei = (const int*)  d_in[1];
    const float* W1 = (const float*)d_in[2];
    const float* b1 = (const float*)d_in[3];
    const float* W2 = (const float*)d_in[4];
    const float* b2 = (const float*)d_in[5];
    const float* W3 = (const float*)d_in[6];
    const float* b3 = (const float*)d_in[7];

    const int N = in_sizes[0] / DF;
    const int E = in_sizes[1] / 2;
    const int* src = ei;
    const int* dst = ei + E;

    float* out = (float*)d_out;

    // Workspace layout: dinv[N] | H[N*128] | AGGa[N*128]   (~103 MB)
    char* ws = (char*)d_ws;
    size_t off0 = ((size_t)N * sizeof(float) + 255) & ~(size_t)255;
    float* dinv = (float*)ws;
    float* H    = (float*)(ws + off0);
    float* AGGa = H + (size_t)N * DF;

    const int strips  = (N + 15) / 16;
    const int gblocks = (strips + 7) / 8;
    const int sblocks = (E + 7) / 8;
    const int nb256   = (N + 255) / 256;
    const int eb256   = (E + 255) / 256;

    // Degree / normalization coefficients (deg computed in-place into dinv).
    k_zero  <<<nb256, 256, 0, stream>>>(dinv, N);
    k_degree<<<eb256, 256, 0, stream>>>(dst, dinv, E);
    k_dinv  <<<nb256, 256, 0, stream>>>(dinv, N);

    // Layer 1: H = x@W1 ; AGGa = H*dinv^2 + b1 ; scatter edges into AGGa.
    k_gemm_wmma<false><<<gblocks, 256, 0, stream>>>(x, W1, b1, dinv, H, AGGa, strips, N);
    k_scatter         <<<sblocks, 256, 0, stream>>>(H, src, dst, dinv, AGGa, E);

    // Layer 2: input relu(AGGa) -> H ; aggregation into d_out (used as AGGb).
    k_gemm_wmma<true> <<<gblocks, 256, 0, stream>>>(AGGa, W2, b2, dinv, H, out, strips, N);
    k_scatter         <<<sblocks, 256, 0, stream>>>(H, src, dst, dinv, out, E);

    // Layer 3: input relu(d_out) -> H ; aggregation into AGGa (no relu after).
    k_gemm_wmma<true> <<<gblocks, 256, 0, stream>>>(out, W3, b3, dinv, H, AGGa, strips, N);
    k_scatter         <<<sblocks, 256, 0, stream>>>(H, src, dst, dinv, AGGa, E);

    // Final L2 row-normalize: AGGa -> d_out.
    k_normalize<<<(N + 7) / 8, 256, 0, stream>>>(AGGa, out, N);
}